// GraphSAGE_43920335569400
// MI455X (gfx1250) — compile-verified
//
#include <hip/hip_runtime.h>
#include <math.h>

// CDNA5 / gfx1250 GraphSAGE. Exact-f32 path: the workload is L2/scatter-bound
// (x fits in the 192MB L2; GEMMs are only 13.1 GFLOP), so we use the exact
// V_WMMA_F32_16X16X4_F32 matrix pipe instead of trading precision for
// throughput we don't need.

typedef __attribute__((ext_vector_type(2))) float v2f;
typedef __attribute__((ext_vector_type(8))) float v8f;

#define DIM 128

// ---------------------------------------------------------------- zero
__global__ void zero_f32(float* __restrict__ p, long n) {
  long i = (long)blockIdx.x * blockDim.x + threadIdx.x;
  long stride = (long)gridDim.x * blockDim.x;
  for (; i < n; i += stride) p[i] = 0.0f;
}

// ---------------------------------------------------- weight re-swizzle
// out = A @ W^T  =>  B[k][n] = W[n][k].
// WMMA f32 16x16x4 B-fragment layout (ISA 7.12.2, mirrors documented A layout):
//   VGPR0 (b.x): lanes 0-15 -> B[k0+0][n0+lane], lanes 16-31 -> B[k0+2][n0+lane-16]
//   VGPR1 (b.y): lanes 0-15 -> B[k0+1][n0+lane], lanes 16-31 -> B[k0+3][n0+lane-16]
// Bbuf index: ((t*64 + j)*32 + lane)*2 floats; j in [0,32) -> Wl, [32,64) -> Wr.
__global__ void reorder_weights(const float* __restrict__ Wl,
                                const float* __restrict__ Wr,
                                float* __restrict__ Bbuf) {
  int id = blockIdx.x * blockDim.x + threadIdx.x;   // 8*64*32 = 16384 threads
  if (id >= 8 * 64 * 32) return;
  int lane = id & 31;
  int j    = (id >> 5) & 63;
  int t    = id >> 11;
  const float* W = (j < 32) ? Wl : Wr;
  int kbase = (j & 31) * 4 + ((lane < 16) ? 0 : 2);
  int n     = t * 16 + (lane & 15);
  Bbuf[id * 2 + 0] = W[n * DIM + kbase + 0];
  Bbuf[id * 2 + 1] = W[n * DIM + kbase + 1];
}

// -------------------------------------------------------- scatter-add
// One wave32 per edge: each lane moves a float4 of the 128-float source row.
// x / agg are L2-resident (51.2 MB each << 192 MB L2), so these atomics and
// gathers stay on-chip.
__global__ void scatter_accum(const float* __restrict__ feat,
                              const int* __restrict__ src,
                              const int* __restrict__ dst,
                              float* __restrict__ agg,
                              float* __restrict__ cnt,
                              int nEdges) {
  int gid  = blockIdx.x * blockDim.x + threadIdx.x;
  int e    = gid >> 5;
  int lane = gid & 31;
  if (e >= nEdges) return;
  int s = src[e];
  int d = dst[e];
  const float4 v = ((const float4*)(feat + (size_t)s * DIM))[lane];
  float* ap = agg + (size_t)d * DIM + lane * 4;
  atomicAdd(ap + 0, v.x);
  atomicAdd(ap + 1, v.y);
  atomicAdd(ap + 2, v.z);
  atomicAdd(ap + 3, v.w);
  if (lane == 0) atomicAdd(cnt + d, 1.0f);
}

// --------------------------------------------------------------- GEMM
// out[m][:] = ELU( (agg[m]/max(cnt,1)) @ Wl^T + b + self[m] @ Wr^T )
// 4 waves / block; each wave computes a 16(M) x 128(N) strip with
// v_wmma_f32_16x16x4_f32, K = 256 (concat of agg-part and self-part).
// A-fragment layout (ISA 7.12.2): v0 = K0/K2 across lane halves, v1 = K1/K3.
__global__ void __launch_bounds__(128)
sage_gemm_elu(const float* __restrict__ aggA,
              const float* __restrict__ selfA,
              const float* __restrict__ cnt,
              const float* __restrict__ Bbuf,
              const float* __restrict__ bias,
              float* __restrict__ out, int nNodes) {
  const int lane = threadIdx.x & 31;
  const int wave = threadIdx.x >> 5;
  const int m0   = blockIdx.x * 64 + wave * 16;

  // A-fragment addressing for this lane (row clamped so EXEC stays all-1s
  // through the WMMAs; clamped rows are never stored).
  const int halfSel = (lane < 16) ? 0 : 2;
  int rowA = m0 + (lane & 15);
  if (rowA > nNodes - 1) rowA = nNodes - 1;
  const float  scale   = 1.0f / fmaxf(cnt[rowA], 1.0f);   // scatter-mean
  const float* aggRow  = aggA  + (size_t)rowA * DIM + halfSel;
  const float* selfRow = selfA + (size_t)rowA * DIM + halfSel;

  v8f acc[8];
#pragma unroll
  for (int t = 0; t < 8; ++t) acc[t] = (v8f){0, 0, 0, 0, 0, 0, 0, 0};

  // --- K-half 1: mean-aggregated neighbors (scale folded into A load) ---
  for (int j = 0; j < 32; ++j) {
    v2f a = *(const v2f*)(aggRow + j * 4);
    a = a * scale;
#pragma unroll
    for (int t = 0; t < 8; ++t) {
      v2f b = *(const v2f*)(Bbuf + ((t * 64 + j) * 32 + lane) * 2);
      acc[t] = __builtin_amdgcn_wmma_f32_16x16x4_f32(
          false, a, false, b, (short)0, acc[t], false, false);
    }
  }
  // --- K-half 2: self features ---
  for (int j = 32; j < 64; ++j) {
    v2f a = *(const v2f*)(selfRow + (j - 32) * 4);
#pragma unroll
    for (int t = 0; t < 8; ++t) {
      v2f b = *(const v2f*)(Bbuf + ((t * 64 + j) * 32 + lane) * 2);
      acc[t] = __builtin_amdgcn_wmma_f32_16x16x4_f32(
          false, a, false, b, (short)0, acc[t], false, false);
    }
  }

  // --- epilogue: bias + ELU(alpha=1), store guarded per row ---
  // C/D layout: VGPR r, lanes 0-15 -> M=r, lanes 16-31 -> M=r+8; N = lane%16.
  const int nBase = lane & 15;
  const int mOff  = (lane < 16) ? 0 : 8;
#pragma unroll
  for (int t = 0; t < 8; ++t) {
    const int   n  = t * 16 + nBase;
    const float bn = bias[n];
#pragma unroll
    for (int r = 0; r < 8; ++r) {
      int row = m0 + mOff + r;
      if (row < nNodes) {
        float v = acc[t][r] + bn;
        v = (v > 0.0f) ? v : expm1f(v);
        out[(size_t)row * DIM + n] = v;
      }
    }
  }
}

// ------------------------------------------------------- log_softmax
// One wave32 per node; float4 per lane covers the 128-wide row; wave32
// __shfl_xor reductions for max and sum-exp. In place.
__global__ void log_softmax_128(float* __restrict__ y, int nNodes) {
  int gid  = blockIdx.x * blockDim.x + threadIdx.x;
  int node = gid >> 5;
  int lane = gid & 31;
  if (node >= nNodes) return;
  float4* row = (float4*)(y + (size_t)node * DIM);
  float4 v = row[lane];
  float m = fmaxf(fmaxf(v.x, v.y), fmaxf(v.z, v.w));
#pragma unroll
  for (int off = 16; off; off >>= 1) m = fmaxf(m, __shfl_xor(m, off, 32));
  float s = expf(v.x - m) + expf(v.y - m) + expf(v.z - m) + expf(v.w - m);
#pragma unroll
  for (int off = 16; off; off >>= 1) s += __shfl_xor(s, off, 32);
  float lse = m + logf(s);
  v.x -= lse; v.y -= lse; v.z -= lse; v.w -= lse;
  row[lane] = v;
}

// --------------------------------------------------------------- launch
extern "C" void kernel_launch(void* const* d_in, const int* in_sizes, int n_in,
                              void* d_out, int out_size, void* d_ws, size_t ws_size,
                              hipStream_t stream) {
  const float* x   = (const float*)d_in[0];
  const int*   ei  = (const int*)d_in[1];   // harness contract: integer -> int32
  const float* W1l = (const float*)d_in[2];
  const float* b1  = (const float*)d_in[3];
  const float* W1r = (const float*)d_in[4];
  const float* W2l = (const float*)d_in[5];
  const float* b2  = (const float*)d_in[6];
  const float* W2r = (const float*)d_in[7];
  float* out = (float*)d_out;

  const int nNodes = in_sizes[0] / DIM;     // 100000
  const int nEdges = in_sizes[1] / 2;       // 600000
  const int* src = ei;
  const int* dst = ei + nEdges;

  // workspace layout (~52 MB): agg | cnt | B1 | B2
  float* agg = (float*)d_ws;                          // nNodes*128
  float* cnt = agg + (size_t)nNodes * DIM;            // nNodes
  float* B1  = cnt + nNodes;                          // 8*64*32*2 = 32768 floats
  float* B2  = B1 + 8 * 64 * 64;

  const long zeroN        = (long)nNodes * DIM + nNodes;  // agg + cnt contiguous
  const int  scatBlocks   = (nEdges * 32 + 255) / 256;
  const int  gemmBlocks   = (nNodes + 63) / 64;
  const int  smaxBlocks   = (nNodes * 32 + 255) / 256;

  // weight fragments (once; hot in L2 for all GEMM blocks)
  reorder_weights<<<64, 256, 0, stream>>>(W1l, W1r, B1);
  reorder_weights<<<64, 256, 0, stream>>>(W2l, W2r, B2);

  // ---- layer 1 ----
  zero_f32<<<1024, 256, 0, stream>>>(agg, zeroN);
  scatter_accum<<<scatBlocks, 256, 0, stream>>>(x, src, dst, agg, cnt, nEdges);
  sage_gemm_elu<<<gemmBlocks, 128, 0, stream>>>(agg, x, cnt, B1, b1, out, nNodes);

  // ---- layer 2 (hidden activations live in d_out; in-place safe: each GEMM
  // block reads only its own rows, all before its epilogue stores) ----
  zero_f32<<<1024, 256, 0, stream>>>(agg, zeroN);
  scatter_accum<<<scatBlocks, 256, 0, stream>>>(out, src, dst, agg, cnt, nEdges);
  sage_gemm_elu<<<gemmBlocks, 128, 0, stream>>>(agg, out, cnt, B2, b2, out, nNodes);

  log_softmax_128<<<smaxBlocks, 256, 0, stream>>>(out, nNodes);

  (void)n_in; (void)out_size; (void)ws_size;
}